// PolicyNetwork_19542101197118
// MI455X (gfx1250) — compile-verified
//
#include <hip/hip_runtime.h>

// ---------------- problem constants ----------------
#define BATCH   64
#define TFRAMES 20
#define KSEL    5
#define CCH     3
#define HRES    224
#define HH      (HRES * HRES)        // 50176
#define LOWD    245760               // 20*3*64*64
#define FEAT    64
#define NPOL    20
#define NEVAL   400

// split-K: 4 M-tile groups * 512 splits = 2048 wave32 blocks
#define NSPLIT  512
#define KCHUNK  (LOWD / NSPLIT)      // 480 = 30 * 16

typedef float v2f __attribute__((ext_vector_type(2)));
typedef float v4f __attribute__((ext_vector_type(4)));
typedef float v8f __attribute__((ext_vector_type(8)));

// ---------------------------------------------------------------------------
// Kernel 1: split-K FP32 WMMA GEMM partials, 16x64 strip per wave.
//   A = Full_Frame64 reshaped [64, 245760] row-major   (streamed once -> NT)
//   B = W_skim [245760, 64] row-major                  (L2-resident, reused)
// Each wave: 1 M-tile x 4 N-tiles => 4 accumulators share each A operand,
// and the 4 B loads per K-row cover the full 256B row (whole cachelines).
// ISA layouts (CDNA5 ISA):
//   A 16x4 f32:  lane = 16*half + m ; VGPR0 = K=2*half, VGPR1 = K=2*half+1
//   B 4x16 f32:  lane = 16*half + n ; VGPR0 = row K=2*half, VGPR1 = K=2*half+1
//   C 16x16 f32: VGPR r: lanes 0-15 -> M=r, lanes 16-31 -> M=r+8, N=lane%16
// ---------------------------------------------------------------------------
__global__ __launch_bounds__(32)
void skim_wmma_partial(const float* __restrict__ A,
                       const float* __restrict__ Bm,
                       float* __restrict__ partials)
{
    const int mtile = blockIdx.x;        // 0..3
    const int split = blockIdx.y;        // 0..NSPLIT-1

    const int lane = threadIdx.x;
    const int half = lane >> 4;          // 0 or 1
    const int lm   = lane & 15;
    const int klo  = half * 2;           // K offset this half-wave owns

    const float* __restrict__ Arow = A  + (long)(mtile * 16 + lm) * LOWD;
    const float* __restrict__ Bcol = Bm + lm;   // +16*ntile per N-tile

    const long kbeg = (long)split * KCHUNK;

    v8f acc0 = {}, acc1 = {}, acc2 = {}, acc3 = {};
#pragma unroll 4
    for (int kk = 0; kk < KCHUNK; kk += 4) {
        const long k = kbeg + kk + klo;

        v2f a;
        a.x = __builtin_nontemporal_load(&Arow[k]);
        a.y = __builtin_nontemporal_load(&Arow[k + 1]);

        const float* __restrict__ r0 = Bcol + k * FEAT;        // B row k
        const float* __restrict__ r1 = Bcol + (k + 1) * FEAT;  // B row k+1
        v2f b0, b1, b2, b3;
        b0.x = r0[0];   b0.y = r1[0];
        b1.x = r0[16];  b1.y = r1[16];
        b2.x = r0[32];  b2.y = r1[32];
        b3.x = r0[48];  b3.y = r1[48];

        acc0 = __builtin_amdgcn_wmma_f32_16x16x4_f32(false, a, false, b0, (short)0, acc0, false, false);
        acc1 = __builtin_amdgcn_wmma_f32_16x16x4_f32(false, a, false, b1, (short)0, acc1, false, false);
        acc2 = __builtin_amdgcn_wmma_f32_16x16x4_f32(false, a, false, b2, (short)0, acc2, false, false);
        acc3 = __builtin_amdgcn_wmma_f32_16x16x4_f32(false, a, false, b3, (short)0, acc3, false, false);
    }

    // store the 4 16x16 tile partials for this (split, mtile) strip
    float* __restrict__ tp = partials + (long)(split * 4 + mtile) * 1024;
    const int ro = 8 * half;
#pragma unroll
    for (int r = 0; r < 8; ++r) {
        const int o = (r + ro) * 16 + lm;
        tp[0 * 256 + o] = acc0[r];
        tp[1 * 256 + o] = acc1[r];
        tp[2 * 256 + o] = acc2[r];
        tp[3 * 256 + o] = acc3[r];
    }
}

// ---------------------------------------------------------------------------
// Kernel 2: deterministic split-K reduction + bias -> fea[64][64]
// ---------------------------------------------------------------------------
__global__ __launch_bounds__(256)
void skim_reduce(const float* __restrict__ partials,
                 const float* __restrict__ b_skim,
                 float* __restrict__ fea)
{
    const int i = blockIdx.x * 256 + threadIdx.x;   // 0..4095
    if (i >= FEAT * FEAT) return;
    const int m = i >> 6, n = i & 63;
    const int mtile = m >> 4, ntile = n >> 4;
    const int off   = ntile * 256 + (m & 15) * 16 + (n & 15);
    float s = 0.0f;
    for (int sp = 0; sp < NSPLIT; ++sp)
        s += partials[(long)(sp * 4 + mtile) * 1024 + off];
    fea[i] = s + b_skim[n];
}

// ---------------------------------------------------------------------------
// Kernel 3: policy head + top-5.  Softmax is monotonic -> top-k on logits.
// Strict '>' argmax passes reproduce top_k's first-occurrence tie-break;
// indices then sorted ascending (matches jnp.sort).
// ---------------------------------------------------------------------------
__global__ __launch_bounds__(32)
void policy_topk(const float* __restrict__ fea,
                 const float* __restrict__ W_pol,
                 const float* __restrict__ b_pol,
                 int* __restrict__ idx)
{
    __shared__ float logits[NPOL];
    const int b = blockIdx.x;
    const int t = threadIdx.x;

    if (t < NPOL) {
        float s = b_pol[t];
        for (int k = 0; k < FEAT; ++k) {
            float f = fea[b * FEAT + k];
            f = f > 0.0f ? f : 0.0f;                 // relu
            s += f * W_pol[k * NPOL + t];
        }
        logits[t] = s;
    }
    __syncthreads();

    if (t == 0) {
        int  sel[KSEL];
        bool used[NPOL] = {};
        for (int j = 0; j < KSEL; ++j) {
            int best = 0; float bv = -INFINITY;
            for (int c = 0; c < NPOL; ++c)
                if (!used[c] && logits[c] > bv) { bv = logits[c]; best = c; }
            used[best] = true;
            sel[j] = best;
        }
        for (int a = 1; a < KSEL; ++a) {             // insertion sort ascending
            int v = sel[a], p = a - 1;
            while (p >= 0 && sel[p] > v) { sel[p + 1] = sel[p]; --p; }
            sel[p + 1] = v;
        }
        for (int j = 0; j < KSEL; ++j) idx[b * KSEL + j] = sel[j];
    }
}

// ---------------------------------------------------------------------------
// Kernel 4: gather selected frames + spatial mean pool.
// One block per (batch, selected-frame, channel): 960 blocks, each streams
// 50176 floats as b128 non-temporal loads (~193 MB total, pure bandwidth,
// single-use data kept out of L2).
// ---------------------------------------------------------------------------
__global__ __launch_bounds__(256)
void gather_pool(const float* __restrict__ F224,
                 const int* __restrict__ idx,
                 float* __restrict__ pooled)
{
    const int blk = blockIdx.x;              // 0..959
    const int b   = blk / (KSEL * CCH);
    const int rem = blk % (KSEL * CCH);
    const int j   = rem / CCH;
    const int cc  = rem % CCH;
    const int fr  = idx[b * KSEL + j];

    const v4f* __restrict__ src =
        (const v4f*)(F224 + ((((long)b * TFRAMES + fr) * CCH + cc) * (long)HH));

    float s = 0.0f;
    for (int i = threadIdx.x; i < HH / 4; i += 256) {   // 12544 v4f's
        v4f v = __builtin_nontemporal_load(&src[i]);
        s += v.x + v.y + v.z + v.w;
    }

    __shared__ float red[256];
    red[threadIdx.x] = s;
    __syncthreads();
    for (int w = 128; w > 0; w >>= 1) {
        if (threadIdx.x < w) red[threadIdx.x] += red[threadIdx.x + w];
        __syncthreads();
    }
    if (threadIdx.x == 0)
        pooled[blk] = red[0] * (1.0f / (float)HH);
}

// ---------------------------------------------------------------------------
// Kernel 5: out = pooled[64,15] @ W_eval[15,400] + b_eval   (25600 tiny dots)
// ---------------------------------------------------------------------------
__global__ __launch_bounds__(256)
void eval_out(const float* __restrict__ pooled,
              const float* __restrict__ W_eval,
              const float* __restrict__ b_eval,
              float* __restrict__ out)
{
    const int i = blockIdx.x * 256 + threadIdx.x;
    if (i >= BATCH * NEVAL) return;
    const int row = i / NEVAL, col = i % NEVAL;
    float s = b_eval[col];
#pragma unroll
    for (int k = 0; k < KSEL * CCH; ++k)
        s += pooled[row * (KSEL * CCH) + k] * W_eval[k * NEVAL + col];
    out[i] = s;
}

// ---------------------------------------------------------------------------
// launch
// ---------------------------------------------------------------------------
extern "C" void kernel_launch(void* const* d_in, const int* in_sizes, int n_in,
                              void* d_out, int out_size, void* d_ws, size_t ws_size,
                              hipStream_t stream)
{
    (void)in_sizes; (void)n_in; (void)out_size; (void)ws_size;

    const float* F64    = (const float*)d_in[0];   // [64,20,3,64,64]
    const float* F224   = (const float*)d_in[1];   // [64,20,3,224,224]
    const float* W_skim = (const float*)d_in[2];   // [245760,64]
    const float* b_skim = (const float*)d_in[3];   // [64]
    const float* W_pol  = (const float*)d_in[4];   // [64,20]
    const float* b_pol  = (const float*)d_in[5];   // [20]
    const float* W_eval = (const float*)d_in[6];   // [15,400]
    const float* b_eval = (const float*)d_in[7];   // [400]
    float*       out    = (float*)d_out;           // [64,400]

    // workspace layout (floats)
    float* wsf      = (float*)d_ws;
    float* fea      = wsf;                 // 4096 floats
    int*   idx      = (int*)(wsf + 4096);  // 320 ints
    float* pooled   = wsf + 4416;          // 960 floats
    float* partials = wsf + 8192;          // 2048 blocks * 1024 floats = 8 MB

    skim_wmma_partial<<<dim3(4, NSPLIT), 32, 0, stream>>>(F64, W_skim, partials);
    skim_reduce      <<<16, 256, 0, stream>>>(partials, b_skim, fea);
    policy_topk      <<<BATCH, 32, 0, stream>>>(fea, W_pol, b_pol, idx);
    gather_pool      <<<BATCH * KSEL * CCH, 256, 0, stream>>>(F224, idx, pooled);
    eval_out         <<<(BATCH * NEVAL + 255) / 256, 256, 0, stream>>>(pooled, W_eval, b_eval, out);
}